// MultiBoxLoss2_86852828659828
// MI455X (gfx1250) — compile-verified
//
#include <hip/hip_runtime.h>

// ---------------------------------------------------------------------------
// MultiBoxLoss (RetinaFace-style) for MI455X / gfx1250.
//   B=64, P=65536, 2 classes. ~523 MB streamed -> 3 scalars => HBM-bound
//   (~22us roofline at 23.3 TB/s). Strategy:
//     * Kernel A: single fused pass. NT loads for the 470MB loc/landm streams
//       (touched once), RT loads for conf/conf_t (50MB, stays in 192MB L2 and
//       is re-read by the mining passes instead of materializing loss_c_prior).
//       Builds per-batch 1024-bin histogram of loss_c_prior (int atomics).
//     * Kernel B: per-batch top-k threshold from histogram (num_neg = min(7*pos, P-1)).
//     * Kernels C1/C2/C3: deterministic tie-break in the threshold bin by prior
//       index (chunk counts -> exclusive scan -> ballot/popc rank, wave32),
//       then masked cross-entropy accumulation.
//     * Kernel R: bitwise-deterministic final reduction of per-block float
//       partials using V_WMMA_F32_16X16X4_F32 with an all-ones B matrix
//       (D = A*1 + C sums 64 values per issue), then finalize the 3 outputs.
// ---------------------------------------------------------------------------

#define NB 64
#define NP 65536
#define NBINS 1024
#define NCHUNK 256       // priors per block / chunks per batch row
#define NBLK (NB * NCHUNK)

// workspace layout (32-bit words)
#define OFF_HIST   0                         // NB*NBINS ints
#define OFF_NUMPOS (OFF_HIST + NB * NBINS)   // NB ints
#define OFF_TBIN   (OFF_NUMPOS + NB)         // NB ints
#define OFF_TR     (OFF_TBIN + NB)           // NB ints
#define OFF_CNT    (OFF_TR + NB)             // 4 ints: N1, N2, Npos
#define OFF_CHUNK  (OFF_CNT + 4)             // NBLK ints
#define OFF_PART   (OFF_CHUNK + NBLK)        // 4 * NBLK floats
#define WS_WORDS   (OFF_PART + 4 * NBLK)

typedef float  vf2 __attribute__((ext_vector_type(2)));
typedef float  vf4 __attribute__((ext_vector_type(4)));
typedef float  v8f __attribute__((ext_vector_type(8)));

__device__ __forceinline__ float wingf(float pr, float tg) {
  // OMEGA=3, EPS=2, C = 3 - 3*ln(2.5)
  float d = fabsf(tg - pr);
  return d < 3.0f ? 3.0f * log1pf(d * 0.5f) : d - 0.2511278043775349f;
}

__device__ __forceinline__ float smooth_l1(float d) {
  d = fabsf(d);
  return d < 1.0f ? 0.5f * d * d : d - 0.5f;
}

// loss_c_prior bin (must be bit-identical in kernels A, C1, C3)
__device__ __forceinline__ int prior_bin(vf2 c, bool pos, float* v_out) {
  float mx  = fmaxf(c.x, c.y);
  float lse = mx + log1pf(__expf(fminf(c.x, c.y) - mx));
  float g   = pos ? c.y : c.x;          // gathered = conf[target_bin]
  float v   = pos ? 0.0f : (lse - g);   // zeroed at positives
  *v_out = v;
  int bin = (v <= 0.0f) ? 0 : (int)(v * 64.0f);   // [0,16) @ 1/64 resolution
  return bin > (NBINS - 1) ? (NBINS - 1) : bin;
}

// ---------------------------------------------------------------------------
// Kernel A: fused wing/smooth-L1 losses + counts + loss_c_prior histogram.
// grid = (NCHUNK, NB), block = 256. One prior per thread.
// ---------------------------------------------------------------------------
__global__ void __launch_bounds__(256) mb_main(
    const float* __restrict__ loc,   const vf2* __restrict__ conf,
    const float* __restrict__ landm, const float* __restrict__ loc_t,
    const float* __restrict__ landm_t, const int* __restrict__ conf_t,
    int* __restrict__ ws_i, float* __restrict__ ws_f) {
  __shared__ int   h[NBINS];
  __shared__ float fred[3][8];
  __shared__ int   ired[3][8];

  const int tid  = threadIdx.x;
  const int lane = tid & 31;
  const int w    = tid >> 5;
  const int b    = blockIdx.y;
  const size_t idx = (size_t)b * NP + blockIdx.x * NCHUNK + tid;

  for (int i = tid; i < NBINS; i += 256) h[i] = 0;
  __syncthreads();

  const int  ct   = conf_t[idx];
  const bool pos1 = ct > 0;     // landmark mask
  const bool pos  = ct != 0;    // localization / classification mask
  const bool mafa = ct == 1;

  // ---- landmark wing loss: 10 floats per prior (NT: streamed once) ----
  const vf2* lp  = reinterpret_cast<const vf2*>(landm)   + idx * 5;
  const vf2* ltp = reinterpret_cast<const vf2*>(landm_t) + idx * 5;
  vf2 a0 = __builtin_nontemporal_load(lp + 0), t0 = __builtin_nontemporal_load(ltp + 0);
  vf2 a1 = __builtin_nontemporal_load(lp + 1), t1 = __builtin_nontemporal_load(ltp + 1);
  vf2 a2 = __builtin_nontemporal_load(lp + 2), t2 = __builtin_nontemporal_load(ltp + 2);
  vf2 a3 = __builtin_nontemporal_load(lp + 3), t3 = __builtin_nontemporal_load(ltp + 3);
  vf2 a4 = __builtin_nontemporal_load(lp + 4), t4 = __builtin_nontemporal_load(ltp + 4);

  float lm = wingf(a0.x, t0.x) + wingf(a0.y, t0.y) +
             wingf(a1.x, t1.x) + wingf(a1.y, t1.y);
  lm = pos1 ? lm : 0.0f;

  // coords 4..9 scaled by {1,1,3,3,3,3}
  float lmm = wingf(a2.x, t2.x) + wingf(a2.y, t2.y) +
              wingf(3.0f * a3.x, 3.0f * t3.x) + wingf(3.0f * a3.y, 3.0f * t3.y) +
              wingf(3.0f * a4.x, 3.0f * t4.x) + wingf(3.0f * a4.y, 3.0f * t4.y);
  lmm = mafa ? lmm : 0.0f;

  // ---- localization smooth-L1: 4+4 floats (NT) ----
  vf4 L = __builtin_nontemporal_load(reinterpret_cast<const vf4*>(loc)   + idx);
  vf4 T = __builtin_nontemporal_load(reinterpret_cast<const vf4*>(loc_t) + idx);
  float ll = smooth_l1(L.x - T.x) + smooth_l1(L.y - T.y) +
             smooth_l1(L.z - T.z) + smooth_l1(L.w - T.w);
  ll = pos ? ll : 0.0f;

  // ---- classification prior loss -> histogram (conf stays L2-resident) ----
  vf2 c = conf[idx];
  float v;
  int bin = prior_bin(c, pos, &v);
  atomicAdd(&h[bin], 1);

  // ---- deterministic block reduction of the three float partials ----
  float vals[3] = {lm, lmm, ll};
#pragma unroll
  for (int q = 0; q < 3; ++q) {
    float s = vals[q];
    s += __shfl_xor(s, 16, 32);
    s += __shfl_xor(s, 8, 32);
    s += __shfl_xor(s, 4, 32);
    s += __shfl_xor(s, 2, 32);
    s += __shfl_xor(s, 1, 32);
    if (lane == 0) fred[q][w] = s;
  }

  // ---- integer counts via wave32 ballot ----
  unsigned long long b1 = __ballot(pos1);
  unsigned long long b2 = __ballot(mafa);
  unsigned long long b3 = __ballot(pos);
  if (lane == 0) {
    ired[0][w] = __popcll(b1);
    ired[1][w] = __popcll(b2);
    ired[2][w] = __popcll(b3);
  }
  __syncthreads();

  if (tid == 0) {
    const int blk = b * NCHUNK + blockIdx.x;
#pragma unroll
    for (int q = 0; q < 3; ++q) {
      float s = 0.0f;
      for (int i = 0; i < 8; ++i) s += fred[q][i];
      ws_f[OFF_PART + q * NBLK + blk] = s;   // ordered, deterministic
    }
    int s1 = 0, s2 = 0, s3 = 0;
    for (int i = 0; i < 8; ++i) { s1 += ired[0][i]; s2 += ired[1][i]; s3 += ired[2][i]; }
    atomicAdd(&ws_i[OFF_CNT + 0], s1);
    atomicAdd(&ws_i[OFF_CNT + 1], s2);
    atomicAdd(&ws_i[OFF_CNT + 2], s3);
    atomicAdd(&ws_i[OFF_NUMPOS + b], s3);
  }
  // flush LDS histogram to the per-batch global histogram
  __syncthreads();
  for (int i = tid; i < NBINS; i += 256)
    if (h[i]) atomicAdd(&ws_i[OFF_HIST + b * NBINS + i], h[i]);
}

// ---------------------------------------------------------------------------
// Kernel B: per-batch top-k threshold from the histogram (1 block, 64 lanes).
// ---------------------------------------------------------------------------
__global__ void mb_thresh(int* __restrict__ ws_i) {
  const int b = threadIdx.x;
  if (b >= NB) return;
  long long np = ws_i[OFF_NUMPOS + b];
  long long nn = 7LL * np;
  if (nn > NP - 1) nn = NP - 1;
  int t = NBINS, R = 0;
  if (nn > 0) {
    long long cum = 0;
    const int* hb = ws_i + OFF_HIST + b * NBINS;
    for (int bin = NBINS - 1; bin >= 0; --bin) {
      int hc = hb[bin];
      if (cum + hc >= nn) { t = bin; R = (int)(nn - cum); break; }
      cum += hc;
    }
  }
  ws_i[OFF_TBIN + b] = t;
  ws_i[OFF_TR + b]   = R;
}

// ---------------------------------------------------------------------------
// Kernel C1: threshold-bin membership count per chunk (L2 hits on conf).
// ---------------------------------------------------------------------------
__global__ void __launch_bounds__(256) mb_chunkcount(
    const vf2* __restrict__ conf, const int* __restrict__ conf_t,
    int* __restrict__ ws_i) {
  __shared__ int wcnt[8];
  const int tid = threadIdx.x, b = blockIdx.y;
  const size_t idx = (size_t)b * NP + blockIdx.x * NCHUNK + tid;
  const int t = ws_i[OFF_TBIN + b];
  const bool pos = conf_t[idx] != 0;
  float v;
  const int bin = prior_bin(conf[idx], pos, &v);
  unsigned long long bal = __ballot(bin == t);
  if ((tid & 31) == 0) wcnt[tid >> 5] = __popcll(bal);
  __syncthreads();
  if (tid == 0) {
    int s = 0;
    for (int i = 0; i < 8; ++i) s += wcnt[i];
    ws_i[OFF_CHUNK + b * NCHUNK + blockIdx.x] = s;
  }
}

// ---------------------------------------------------------------------------
// Kernel C2: per-batch exclusive scan of the 256 chunk counts.
// ---------------------------------------------------------------------------
__global__ void __launch_bounds__(256) mb_scan(int* __restrict__ ws_i) {
  __shared__ int s[NCHUNK];
  const int b = blockIdx.x, tid = threadIdx.x;
  int* row = ws_i + OFF_CHUNK + b * NCHUNK;
  const int v = row[tid];
  s[tid] = v;
  __syncthreads();
  for (int off = 1; off < NCHUNK; off <<= 1) {
    int x = (tid >= off) ? s[tid - off] : 0;
    __syncthreads();
    if (tid >= off) s[tid] += x;
    __syncthreads();
  }
  row[tid] = s[tid] - v;   // exclusive
}

// ---------------------------------------------------------------------------
// Kernel C3: select pos|neg (deterministic index-order tie-break inside the
// threshold bin), accumulate adjusted cross-entropy, per-block partials.
// ---------------------------------------------------------------------------
__global__ void __launch_bounds__(256) mb_ce(
    const vf2* __restrict__ conf, const int* __restrict__ conf_t,
    const int* __restrict__ ws_i, float* __restrict__ ws_f) {
  __shared__ int   wcnt[8];
  __shared__ float fsum[8];
  const int tid = threadIdx.x, lane = tid & 31, w = tid >> 5, b = blockIdx.y;
  const size_t idx = (size_t)b * NP + blockIdx.x * NCHUNK + tid;
  const int t = ws_i[OFF_TBIN + b];
  const int R = ws_i[OFF_TR + b];
  const int base = ws_i[OFF_CHUNK + b * NCHUNK + blockIdx.x];

  const bool pos = conf_t[idx] != 0;
  const vf2 c = conf[idx];
  float v;
  const int bin = prior_bin(c, pos, &v);

  const bool inT = (bin == t);
  unsigned long long bal = __ballot(inT);
  const int lanePfx = __popcll(bal & ((1ULL << lane) - 1ULL));
  if (lane == 0) wcnt[w] = __popcll(bal);
  __syncthreads();
  int wbase = 0;
  for (int i = 0; i < w; ++i) wbase += wcnt[i];
  const int rank = base + wbase + lanePfx;

  const bool neg = (bin > t) || (inT && rank < R);
  const bool sel = pos || neg;

  // adjusted CE: conf[:,1] -= 0.1 where target==1, then lse - adj[target]
  const float a0 = c.x;
  const float a1 = c.y - (pos ? 0.1f : 0.0f);
  const float mx = fmaxf(a0, a1);
  const float lse = mx + log1pf(__expf(fminf(a0, a1) - mx));
  const float ce = lse - (pos ? a1 : a0);
  float s = sel ? ce : 0.0f;

  s += __shfl_xor(s, 16, 32);
  s += __shfl_xor(s, 8, 32);
  s += __shfl_xor(s, 4, 32);
  s += __shfl_xor(s, 2, 32);
  s += __shfl_xor(s, 1, 32);
  if (lane == 0) fsum[w] = s;
  __syncthreads();
  if (tid == 0) {
    float tot = 0.0f;
    for (int i = 0; i < 8; ++i) tot += fsum[i];
    ws_f[OFF_PART + 3 * NBLK + b * NCHUNK + blockIdx.x] = tot;
  }
}

// ---------------------------------------------------------------------------
// Kernel R: deterministic WMMA-based reduction of the 4 partial arrays + final
// output. Each of the 4 wave32s reduces one 16384-float array with
// V_WMMA_F32_16X16X4_F32 and an all-ones B: D = A*1 + C sums 64 values/issue.
// ---------------------------------------------------------------------------
__global__ void __launch_bounds__(128) mb_reduce(
    const float* __restrict__ part, const int* __restrict__ cnt,
    float* __restrict__ out) {
  __shared__ float red[4];
  const int lane = threadIdx.x & 31;
  const int w    = threadIdx.x >> 5;          // wave -> which array
  const float* arr = part + w * NBLK;

  v8f c;
#pragma unroll
  for (int i = 0; i < 8; ++i) c[i] = 0.0f;
  vf2 ones; ones.x = 1.0f; ones.y = 1.0f;

  // A (16x4 f32): lanes 0-15 hold K={0,1} of row M=lane, lanes 16-31 K={2,3}.
  const int m     = lane & 15;
  const int kbase = (lane >> 4) * 2;
  for (int i = 0; i < NBLK; i += 64) {
    vf2 a;
    a.x = arr[i + m * 4 + kbase + 0];
    a.y = arr[i + m * 4 + kbase + 1];
    c = __builtin_amdgcn_wmma_f32_16x16x4_f32(
        /*neg_a=*/false, a, /*neg_b=*/false, ones,
        /*c_mod=*/(short)0, c, /*reuse_a=*/false, /*reuse_b=*/false);
  }
  // column 0 lives in lanes 0 (rows 0-7) and 16 (rows 8-15)
  float s = c[0] + c[1] + c[2] + c[3] + c[4] + c[5] + c[6] + c[7];
  s += __shfl_xor(s, 16, 32);
  if (lane == 0) red[w] = s;
  __syncthreads();

  if (threadIdx.x == 0) {
    const float loss_landm = red[0];
    const float loss_mafa  = red[1];
    const float loss_l     = red[2];
    const float loss_c     = red[3];
    const float N1 = fmaxf((float)cnt[0], 1.0f);
    const float N2 = fmaxf((float)cnt[1], 1.0f);
    const float N  = fmaxf((float)cnt[2], 1.0f);
    out[0] = loss_l / N;
    out[1] = loss_c / N;
    out[2] = loss_landm / N1 + loss_mafa / N2;
  }
}

// ---------------------------------------------------------------------------
extern "C" void kernel_launch(void* const* d_in, const int* in_sizes, int n_in,
                              void* d_out, int out_size, void* d_ws, size_t ws_size,
                              hipStream_t stream) {
  (void)in_sizes; (void)n_in; (void)out_size; (void)ws_size;
  const float* loc     = (const float*)d_in[0];
  const vf2*   conf    = (const vf2*)  d_in[1];
  const float* landm   = (const float*)d_in[2];
  const float* loc_t   = (const float*)d_in[3];
  const float* landm_t = (const float*)d_in[4];
  const int*   conf_t  = (const int*)  d_in[5];
  float* out = (float*)d_out;
  int*   wsi = (int*)d_ws;
  float* wsf = (float*)d_ws;

  // zero hist / counters / thresholds (partials & chunk counts fully rewritten)
  hipMemsetAsync(d_ws, 0, (size_t)OFF_CHUNK * sizeof(int), stream);

  dim3 blk(256), grd(NCHUNK, NB);
  mb_main<<<grd, blk, 0, stream>>>(loc, conf, landm, loc_t, landm_t, conf_t, wsi, wsf);
  mb_thresh<<<1, 64, 0, stream>>>(wsi);
  mb_chunkcount<<<grd, blk, 0, stream>>>(conf, conf_t, wsi);
  mb_scan<<<NB, NCHUNK, 0, stream>>>(wsi);
  mb_ce<<<grd, blk, 0, stream>>>(conf, conf_t, wsi, wsf);
  mb_reduce<<<1, 128, 0, stream>>>(wsf + OFF_PART, wsi + OFF_CNT, out);
}